// PC_shielded_electrostatics_36859409334420
// MI455X (gfx1250) — compile-verified
//
#include <hip/hip_runtime.h>
#include <hip/hip_bf16.h>

typedef float v4f __attribute__((ext_vector_type(4)));
typedef int   v4i __attribute__((ext_vector_type(4)));

#define SHORT_CUT   4.0f
#define INV_SHORT   0.25f
#define LONG_CUT    10.0f
#define INV_LR2     0.01f      // 1 / (LONG_CUT^2)
#define TWO_OVER_LC 0.2f       // 2 / LONG_CUT
#define KEHALF      7.199822675975274f

__global__ void zero_out_kernel(float* __restrict__ out, int n) {
    int t = blockIdx.x * blockDim.x + threadIdx.x;
    if (t < n) out[t] = 0.0f;
}

// Fast 1-ULP reciprocal / sqrt: single v_rcp_f32 / v_sqrt_f32, no Newton
// refinement. Inputs here are in [0.5, 12] -> no denormal/inf edge cases.
__device__ __forceinline__ float fast_rcp(float x) {
#if defined(__has_builtin) && __has_builtin(__builtin_amdgcn_rcpf)
    return __builtin_amdgcn_rcpf(x);
#else
    return 1.0f / x;
#endif
}
__device__ __forceinline__ float fast_sqrt(float x) {
#if defined(__has_builtin) && __has_builtin(__builtin_amdgcn_sqrtf)
    return __builtin_amdgcn_sqrtf(x);
#else
    return sqrtf(x);
#endif
}

__device__ __forceinline__ float edge_energy(float d, float qi, float qj) {
    // shielded distance
    float d_sh = fast_sqrt(__builtin_fmaf(d, d, 1.0f));
    // smooth switch 6x^5 - 15x^4 + 10x^3, clamped to 1 for d >= SHORT_CUT
    float x  = d * INV_SHORT;
    float x3 = x * x * x;
    float sw = ((6.0f * x - 15.0f) * x + 10.0f) * x3;
    sw = (d < SHORT_CUT) ? sw : 1.0f;
    float csw = 1.0f - sw;
    // damped-shifted Coulomb terms
    float E_ord = fast_rcp(d)    + __builtin_fmaf(d,    INV_LR2, -TWO_OVER_LC);
    float E_shd = fast_rcp(d_sh) + __builtin_fmaf(d_sh, INV_LR2, -TWO_OVER_LC);
    return KEHALF * (qi * qj) * __builtin_fmaf(csw, E_shd, sw * E_ord);
}

__global__ void __launch_bounds__(256)
PC_shielded_electrostatics_kernel(const float* __restrict__ q,
                                  const float* __restrict__ dist,
                                  const int*   __restrict__ idx_i,
                                  const int*   __restrict__ idx_j,
                                  float*       __restrict__ out,
                                  int n_edges) {
    int t    = blockIdx.x * blockDim.x + threadIdx.x;
    int base = t * 4;

    if (base + 4 <= n_edges) {
        // Hot path: three B128 non-temporal streaming loads cover 4 edges.
        v4f d4 = __builtin_nontemporal_load((const v4f*)(dist)  + t);
        v4i i4 = __builtin_nontemporal_load((const v4i*)(idx_i) + t);
        v4i j4 = __builtin_nontemporal_load((const v4i*)(idx_j) + t);
#pragma unroll
        for (int k = 0; k < 4; ++k) {
            float d = d4[k];
            if (d <= LONG_CUT) {               // zero contribution beyond cutoff
                int   ii = i4[k];
                float qi = q[ii];              // 2MB table: L2/WGP$ resident
                float qj = q[j4[k]];
                float E  = edge_energy(d, qi, qj);
                unsafeAtomicAdd(&out[ii], E);  // global_atomic_add_f32, no-return
            }
        }
    } else if (base < n_edges) {
        // Tail (not hit for N_EDGES = 32M, kept for generality)
        for (int e = base; e < n_edges; ++e) {
            float d = dist[e];
            if (d <= LONG_CUT) {
                int   ii = idx_i[e];
                float E  = edge_energy(d, q[ii], q[idx_j[e]]);
                unsafeAtomicAdd(&out[ii], E);
            }
        }
    }
}

extern "C" void kernel_launch(void* const* d_in, const int* in_sizes, int n_in,
                              void* d_out, int out_size, void* d_ws, size_t ws_size,
                              hipStream_t stream) {
    const float* q     = (const float*)d_in[0];
    const float* dist  = (const float*)d_in[1];
    const int*   idx_i = (const int*)  d_in[2];
    const int*   idx_j = (const int*)  d_in[3];
    float*       out   = (float*)d_out;

    const int n_atoms = in_sizes[0];
    const int n_edges = in_sizes[1];

    // 1) zero the scatter target (harness poisons d_out with 0xAA)
    {
        int threads = 256;
        int blocks  = (n_atoms + threads - 1) / threads;
        zero_out_kernel<<<blocks, threads, 0, stream>>>(out, n_atoms);
    }

    // 2) edge kernel: 4 edges per lane (B128 streaming loads) + L2 fp atomics
    {
        int threads = 256;
        int n_vec   = (n_edges + 3) / 4;
        int blocks  = (n_vec + threads - 1) / threads;
        PC_shielded_electrostatics_kernel<<<blocks, threads, 0, stream>>>(
            q, dist, idx_i, idx_j, out, n_edges);
    }
}